// ImagePatchFilter_66812511257257
// MI455X (gfx1250) — compile-verified
//
#include <hip/hip_runtime.h>
#include <stdint.h>

// ---------------------------------------------------------------------------
// ImagePatchFilter for MI455X (gfx1250), compile-only tuning.
//
// Memory-bound op (~428 MB total HBM traffic -> ~18us @ 23.3 TB/s). No matrix
// math -> WMMA not applicable; the CDNA5 features used are the Tensor Data
// Mover (tensor_load_to_lds + s_wait_tensorcnt) for bulk input staging,
// wave32 shuffles, LDS bitonic select, and non-temporal output stores so the
// 201 MB write stream does not evict the input images from the 192 MB L2
// (kernel C re-reads ~12.5% of the input).
// ---------------------------------------------------------------------------

#define NCH    3
#define HW     512
#define PATCH  8
#define NPH    64      // patch rows/cols per image
#define PPI    4096    // patches per image
#define TOPK   512
#define EPSV   1e-8f

typedef unsigned int u32x4 __attribute__((ext_vector_type(4)));
typedef int          i32x4 __attribute__((ext_vector_type(4)));
typedef int          i32x8 __attribute__((ext_vector_type(8)));
typedef float        f32x4 __attribute__((ext_vector_type(4)));

// ---------------------------------------------------------------------------
// Kernel A: one block per (image b, patch-row ph). TDM loads the 3x8x512 f32
// slab (48 KB) into LDS, then 4 lanes per patch reduce max/min and emit the
// contrast scalar.
// ---------------------------------------------------------------------------
__global__ __launch_bounds__(256) void patch_contrast_kernel(
    const float* __restrict__ images, float* __restrict__ contrast) {
  __shared__ float tile[NCH * PATCH * HW];  // 48 KB, layout [(c*8+i)*512 + w]

  const unsigned bid = blockIdx.x;
  const unsigned b   = bid >> 6;    // image
  const unsigned ph  = bid & 63u;   // patch row

  // ---- TDM: issue from wave 0 only (uniform branch; TDM ignores EXEC, so
  // gate at wave granularity, not lane granularity). ----
  if (threadIdx.x < 32u) {
    const uint64_t gaddr = (uint64_t)(uintptr_t)(
        images + (size_t)b * NCH * HW * HW + (size_t)ph * PATCH * HW);
    const unsigned ldsAddr = (unsigned)(uintptr_t)(&tile[0]);  // low 32b = LDS offset

    // D# group 0 (ISA 08 §8.3): count=1 | lds_addr | global_addr[56:0] | type=2
    u32x4 g0 = { 1u,
                 ldsAddr,
                 (unsigned)(gaddr & 0xFFFFFFFFull),
                 (unsigned)(((gaddr >> 32) & 0x01FFFFFFull) | 0x80000000u) };
    // D# group 1 (§8.4): data_size=4B; tensor_dim0=512, tensor_dim1=512;
    // tile = 512 x 8 x 3; stride0 = 512 (row), stride1 = 512*512 (channel).
    i32x8 g1 = { (int)(2u << 16),                 // data_size=2 (4 bytes)
                 (int)(HW << 16),                 // tensor_dim0[15:0] in [31:16]
                 (int)(HW << 16),                 // dim0 hi=0 | tensor_dim1 lo=512
                 (int)(HW << 16),                 // dim1 hi=0 | tile_dim0=512
                 (int)(PATCH | (NCH << 16)),      // tile_dim1=8 | tile_dim2=3
                 (int)HW,                         // tensor_dim0_stride lo32 = 512
                 0,                               // stride0 hi | stride1 lo16 (0x40000 lo16 = 0)
                 (int)((HW * HW) >> 16) };        // tensor_dim1_stride[47:16] = 4
    // D# group 2 (§8.5): tensor_dim2=3; no dim3, no iterate.
    i32x4 g2 = { NCH, 0, 0, 0 };
    i32x4 g3 = { 0, 0, 0, 0 };
    // Trailing group (6-arg toolchain form): unused extended state, zeros.
    i32x8 g4 = { 0, 0, 0, 0, 0, 0, 0, 0 };

    __builtin_amdgcn_tensor_load_to_lds(g0, g1, g2, g3, g4, /*cpol=*/0);
    __builtin_amdgcn_s_wait_tensorcnt(0);
  }
  __syncthreads();

  // ---- Reduce: 4 threads per patch, 6 rows of 8 floats each. ----
  const unsigned pw  = threadIdx.x >> 2;   // patch column 0..63
  const unsigned sub = threadIdx.x & 3u;   // sub-reducer 0..3
  float mx = -3.0e38f, mn = 3.0e38f;
#pragma unroll
  for (int r = 0; r < 6; ++r) {
    const float* row = &tile[(sub * 6u + r) * HW + pw * PATCH];
    f32x4 a = *(const f32x4*)(row);
    f32x4 c = *(const f32x4*)(row + 4);
    mx = fmaxf(mx, fmaxf(fmaxf(a.x, a.y), fmaxf(a.z, a.w)));
    mx = fmaxf(mx, fmaxf(fmaxf(c.x, c.y), fmaxf(c.z, c.w)));
    mn = fminf(mn, fminf(fminf(a.x, a.y), fminf(a.z, a.w)));
    mn = fminf(mn, fminf(fminf(c.x, c.y), fminf(c.z, c.w)));
  }
  // wave32 shuffle reduce across the 4 lanes of this patch
  mx = fmaxf(mx, __shfl_xor(mx, 1));
  mx = fmaxf(mx, __shfl_xor(mx, 2));
  mn = fminf(mn, __shfl_xor(mn, 1));
  mn = fminf(mn, __shfl_xor(mn, 2));

  if (sub == 0) {
    contrast[(size_t)bid * 64u + pw] = (mx - mn + EPSV) / (mx + mn);
  }
}

// ---------------------------------------------------------------------------
// Kernel B: one block per image. Bitonic-sort 4096 unique 64-bit keys
// (value bits desc, index asc) in LDS; key[511] is the selection threshold.
// ---------------------------------------------------------------------------
__global__ __launch_bounds__(512) void topk_threshold_kernel(
    const float* __restrict__ contrast, unsigned long long* __restrict__ thresh) {
  __shared__ unsigned long long keys[PPI];  // 32 KB
  const unsigned b = blockIdx.x;

  for (unsigned p = threadIdx.x; p < PPI; p += 512u) {
    const unsigned bits = __float_as_uint(contrast[(size_t)b * PPI + p]);
    // contrast >= 0 -> IEEE bit pattern is order-preserving as unsigned.
    keys[p] = ((unsigned long long)bits << 32) |
              (unsigned long long)(0xFFFFFFFFu - p);  // lower index wins ties
  }
  __syncthreads();

  // Bitonic sort, descending.
  for (unsigned k = 2; k <= PPI; k <<= 1) {
    for (unsigned j = k >> 1; j > 0; j >>= 1) {
      for (unsigned t = threadIdx.x; t < (PPI / 2); t += 512u) {
        const unsigned i   = 2u * t - (t & (j - 1u));
        const unsigned ixj = i + j;
        const bool descBlock = ((i & k) == 0u);
        const unsigned long long a = keys[i];
        const unsigned long long c = keys[ixj];
        const bool sw = descBlock ? (a < c) : (a > c);
        if (sw) { keys[i] = c; keys[ixj] = a; }
      }
      __syncthreads();
    }
  }
  if (threadIdx.x == 0) thresh[b] = keys[TOPK - 1];  // 512th-largest key
}

// ---------------------------------------------------------------------------
// Kernel C: one block per (b, ph). Rebuild keys for 64 patches, compare to
// threshold, then stream the contiguous 48 KB output chunk as non-temporal
// float4 stores (output is never re-read; keep L2 for the input), reading
// input only for kept patches (~12.5%, expected to hit L2).
// ---------------------------------------------------------------------------
__global__ __launch_bounds__(256) void mask_scatter_kernel(
    const float* __restrict__ images, const float* __restrict__ contrast,
    const unsigned long long* __restrict__ thresh, float* __restrict__ out) {
  __shared__ unsigned keep[NPH];
  const unsigned bid = blockIdx.x;
  const unsigned b   = bid >> 6;
  const unsigned ph  = bid & 63u;

  if (threadIdx.x < NPH) {
    const unsigned p   = ph * 64u + threadIdx.x;
    const unsigned bit = __float_as_uint(contrast[(size_t)bid * 64u + threadIdx.x]);
    const unsigned long long key =
        ((unsigned long long)bit << 32) | (unsigned long long)(0xFFFFFFFFu - p);
    keep[threadIdx.x] = (key >= thresh[b]) ? 1u : 0u;  // keys unique -> exact top-512
  }
  __syncthreads();

  // Output chunk for this block: 64 patches * 192 floats = 3072 float4,
  // contiguous and 16B-aligned -> perfectly coalesced NT stores.
  f32x4* out4 = (f32x4*)out + (size_t)bid * 3072u;
  for (unsigned v = threadIdx.x; v < 3072u; v += 256u) {
    const unsigned pw = v / 48u;          // patch column
    const unsigned r  = v % 48u;          // float4 index within patch
    const unsigned c  = r >> 4;           // channel
    const unsigned rr = r & 15u;
    const unsigned i  = rr >> 1;          // pixel row
    const unsigned j4 = (rr & 1u) << 2;   // pixel col (float4 granularity)
    f32x4 val = { 0.f, 0.f, 0.f, 0.f };
    if (keep[pw]) {
      const size_t inOff =
          ((size_t)(b * NCH + c) * HW + (size_t)(ph * PATCH + i)) * HW +
          (size_t)pw * PATCH + j4;
      val = *(const f32x4*)(images + inOff);
    }
    __builtin_nontemporal_store(val, &out4[v]);
  }
}

// ---------------------------------------------------------------------------
extern "C" void kernel_launch(void* const* d_in, const int* in_sizes, int n_in,
                              void* d_out, int out_size, void* d_ws, size_t ws_size,
                              hipStream_t stream) {
  (void)n_in; (void)out_size; (void)ws_size;
  const float* images = (const float*)d_in[0];
  float* out = (float*)d_out;

  const int B = in_sizes[0] / (NCH * HW * HW);  // 64 for the reference shapes

  // Workspace layout: [contrast: B*4096 f32][thresh: B u64]
  float* contrast = (float*)d_ws;
  unsigned long long* thresh =
      (unsigned long long*)((char*)d_ws + (size_t)B * PPI * sizeof(float));

  patch_contrast_kernel<<<B * NPH, 256, 0, stream>>>(images, contrast);
  topk_threshold_kernel<<<B, 512, 0, stream>>>(contrast, thresh);
  mask_scatter_kernel<<<B * NPH, 256, 0, stream>>>(images, contrast, thresh, out);
}